// HDClassifier_48103633715288
// MI455X (gfx1250) — compile-verified
//
#include <hip/hip_runtime.h>
#include <hip/hip_bf16.h>

// ---------------------------------------------------------------------------
// HD classifier for MI455X (gfx1250, wave32).
//   encode: gather/bind/bundle/ngram  -> sign vector per batch (L2-resident,
//           LDS ring buffer for the 4-deep temporal window with D-halo of 3)
//   score : [32,10000] x [10000,6] exact-fp32 GEMM via V_WMMA_F32_16X16X4_F32
// ---------------------------------------------------------------------------

typedef __attribute__((ext_vector_type(2))) float v2f;
typedef __attribute__((ext_vector_type(8))) float v8f;

constexpr int B_    = 32;
constexpr int T_    = 128;
constexpr int C_    = 8;
constexpr int D_    = 10000;
constexpr int NLEV  = 201;
constexpr int NCLS  = 6;

constexpr int TILE    = 512;           // output d-elements per block
constexpr int HALO    = 3;             // ngram needs d-1..d-3 (mod D)
constexpr int THREADS = 256;           // 8 waves of 32
constexpr int NITER   = 3;             // ceil((TILE+HALO)/THREADS)
constexpr int DT      = (D_ + TILE - 1) / TILE;   // 20 tiles

// ---------------------------------------------------------------------------
// Kernel 1: encode.  One block per (d-tile, batch).
// SCORES=false: write hard-quantized sign vector to `signs` [B][D].
// SCORES=true : fused fallback, accumulate class scores via block reduction
//               + atomicAdd (used only if workspace is too small).
// ---------------------------------------------------------------------------
template <bool SCORES>
__global__ __launch_bounds__(THREADS)
void hd_encode(const float* __restrict__ x,
               const float* __restrict__ lvl,     // [NLEV][D] +/-1
               const float* __restrict__ chv,     // [C][D]    +/-1
               const float* __restrict__ cent,    // [NCLS][D] (fallback only)
               float* __restrict__ signs,         // [B][D]    (primary path)
               float* __restrict__ out)           // [B][NCLS] (fallback only)
{
    __shared__ int   s_off[T_ * C_];              // level row byte-free offsets (idx*D)
    __shared__ float ring[4][TILE + HALO];        // 4-deep sliding window of bundled rows

    const int tid    = threadIdx.x;
    const int b      = blockIdx.y;
    const int start  = blockIdx.x * TILE;
    const int tile_n = (D_ - start < TILE) ? (D_ - start) : TILE;
    const int Lloc   = tile_n + HALO;             // local elements incl. halo

    // --- quantize x[b] -> level-row offsets (jnp.round == round-half-even) ---
    const float* xb = x + (size_t)b * T_ * C_;
    for (int i = tid; i < T_ * C_; i += THREADS) {
        // (x - LOW)/(HIGH-LOW)*(NLEV-1) == x + 100 for these constants
        float f = rintf((xb[i] + 100.0f) * ((float)(NLEV - 1) / 200.0f));
        f = fminf(fmaxf(f, 0.0f), (float)(NLEV - 1));
        s_off[i] = (int)f * D_;
    }
    __syncthreads();

    // --- per-thread assignment: local elements ld, global d with wraparound ---
    int   ld[NITER];
    int   gd[NITER];
    bool  valid[NITER];
    float ch[NITER][C_];
    float acc[NITER];
    #pragma unroll
    for (int j = 0; j < NITER; j++) {
        ld[j]    = tid + j * THREADS;
        valid[j] = ld[j] < Lloc;
        int g = start - HALO + ld[j];
        if (g < 0) g += D_;                       // circular halo (tile 0)
        gd[j]  = valid[j] ? g : 0;
        acc[j] = 0.0f;
        #pragma unroll
        for (int c = 0; c < C_; c++)              // channel HV: constant over t
            ch[j][c] = chv[c * D_ + gd[j]];
    }

    // --- temporal loop: bundle channels, 4-gram product via LDS ring ---
    for (int t = 0; t < T_; t++) {
        int ro[C_];
        #pragma unroll
        for (int c = 0; c < C_; c++) ro[c] = s_off[t * C_ + c];   // uniform

        float* rw = ring[t & 3];
        #pragma unroll
        for (int j = 0; j < NITER; j++) {
            if (valid[j]) {
                float s = 0.0f;
                #pragma unroll
                for (int c = 0; c < C_; c++)       // coalesced L2-resident gathers
                    s = fmaf(lvl[(size_t)ro[c] + gd[j]], ch[j][c], s);
                rw[ld[j]] = s;
            }
        }
        __syncthreads();

        if (t >= 3) {
            const float* r0 = ring[(t - 3) & 3];
            const float* r1 = ring[(t - 2) & 3];
            const float* r2 = ring[(t - 1) & 3];
            const float* r3 = ring[(t - 0) & 3];
            #pragma unroll
            for (int j = 0; j < NITER; j++) {
                if (valid[j] && ld[j] >= HALO) {
                    // gram[w,d] = bun[w,d-3]*bun[w+1,d-2]*bun[w+2,d-1]*bun[w+3,d]
                    acc[j] += r0[ld[j] - 3] * r1[ld[j] - 2] *
                              r2[ld[j] - 1] * r3[ld[j]];
                }
            }
        }
        __syncthreads();   // protect ring[(t+1)&3] from being overwritten early
    }

    if (!SCORES) {
        #pragma unroll
        for (int j = 0; j < NITER; j++) {
            if (valid[j] && ld[j] >= HALO) {
                int d = start + ld[j] - HALO;
                signs[(size_t)b * D_ + d] = (acc[j] > 0.0f) ? 1.0f : -1.0f;
            }
        }
    } else {
        float p[NCLS] = {0, 0, 0, 0, 0, 0};
        #pragma unroll
        for (int j = 0; j < NITER; j++) {
            if (valid[j] && ld[j] >= HALO) {
                int d   = start + ld[j] - HALO;
                float s = (acc[j] > 0.0f) ? 1.0f : -1.0f;
                #pragma unroll
                for (int k = 0; k < NCLS; k++) p[k] += s * cent[k * D_ + d];
            }
        }
        __syncthreads();
        float* red = &ring[0][0];                 // 2060 floats >= 6*256
        #pragma unroll
        for (int k = 0; k < NCLS; k++) red[k * THREADS + tid] = p[k];
        __syncthreads();
        for (int s = THREADS / 2; s > 0; s >>= 1) {
            if (tid < s) {
                #pragma unroll
                for (int k = 0; k < NCLS; k++)
                    red[k * THREADS + tid] += red[k * THREADS + tid + s];
            }
            __syncthreads();
        }
        if (tid < NCLS) atomicAdd(&out[b * NCLS + tid], red[tid * THREADS]);
    }
}

// ---------------------------------------------------------------------------
// Kernel 2: out[32,6] = signs[32,10000] @ centroid.T  via V_WMMA_F32_16X16X4_F32
// 1 block, 8 waves. Wave group (rows 0-15 / 16-31) x 4-way K split,
// LDS reduction of the 4 partial accumulators per row group.
// A 16x4 layout (ISA 7.12.2): lane m=lane&15, VGPR0/1 = K = 2*(lane>>4)+{0,1}.
// B 4x16 mirrors with N = lane&15.  C/D: VGPR r -> M = r + 8*(lane>>4), N = lane&15.
// ---------------------------------------------------------------------------
__global__ __launch_bounds__(256)
void hd_score_wmma(const float* __restrict__ signs,
                   const float* __restrict__ cent,
                   float* __restrict__ out)
{
    __shared__ float red[8][32][8];

    const int tid    = threadIdx.x;
    const int wave   = tid >> 5;       // 0..7
    const int lane   = tid & 31;
    const int grp    = wave >> 2;      // 0: rows 0-15, 1: rows 16-31
    const int kslice = wave & 3;       // K split
    const int m0     = grp * 16;
    const int hi     = lane >> 4;
    const int mn     = lane & 15;

    const float* arow = signs + (size_t)(m0 + mn) * D_;
    const float* brow = cent  + (size_t)(mn < NCLS ? mn : 0) * D_;
    const bool   bval = (mn < NCLS);   // pad classes 6..15 with zeros

    v8f acc = {};
    const int k0 = kslice * (D_ / 4);  // 2500-element K chunk, multiple of 4
    for (int k = k0; k < k0 + D_ / 4; k += 4) {
        const int kk = k + 2 * hi;
        v2f a, bb;
        a.x  = arow[kk];
        a.y  = arow[kk + 1];
        float b0 = brow[kk], b1 = brow[kk + 1];
        bb.x = bval ? b0 : 0.0f;
        bb.y = bval ? b1 : 0.0f;
        // D = A(16x4) * B(4x16) + C, exact fp32 — matches reference math
        acc = __builtin_amdgcn_wmma_f32_16x16x4_f32(
                  false, a, false, bb, (short)0, acc, false, false);
    }

    #pragma unroll
    for (int r = 0; r < 8; r++) red[wave][lane][r] = acc[r];
    __syncthreads();

    if (kslice == 0 && mn < NCLS) {
        #pragma unroll
        for (int r = 0; r < 8; r++) {
            float s = red[grp * 4 + 0][lane][r] + red[grp * 4 + 1][lane][r] +
                      red[grp * 4 + 2][lane][r] + red[grp * 4 + 3][lane][r];
            const int m = m0 + r + 8 * hi;
            out[m * NCLS + mn] = s;
        }
    }
}

// ---------------------------------------------------------------------------
extern "C" void kernel_launch(void* const* d_in, const int* in_sizes, int n_in,
                              void* d_out, int out_size, void* d_ws, size_t ws_size,
                              hipStream_t stream)
{
    const float* x    = (const float*)d_in[0];   // [32,128,8]
    const float* lvl  = (const float*)d_in[1];   // [201,10000]
    const float* chv  = (const float*)d_in[2];   // [8,10000]
    const float* cent = (const float*)d_in[3];   // [6,10000]
    float*       out  = (float*)d_out;           // [32,6]

    const size_t need = (size_t)B_ * D_ * sizeof(float);
    dim3 grid1(DT, B_);

    if (ws_size >= need) {
        float* signs = (float*)d_ws;
        hd_encode<false><<<grid1, THREADS, 0, stream>>>(x, lvl, chv, nullptr,
                                                        signs, nullptr);
        hd_score_wmma<<<1, 256, 0, stream>>>(signs, cent, out);
    } else {
        hipMemsetAsync(out, 0, (size_t)out_size * sizeof(float), stream);
        hd_encode<true><<<grid1, THREADS, 0, stream>>>(x, lvl, chv, cent,
                                                       nullptr, out);
    }
}